// Convolution_12936441496323
// MI455X (gfx1250) — compile-verified
//
#include <hip/hip_runtime.h>
#include <math.h>

#define N_NODES 10000
#define N_EDGES 320000

typedef __attribute__((ext_vector_type(2))) float v2f;
typedef __attribute__((ext_vector_type(8))) float v8f;

__device__ __forceinline__ v8f wmma4(v2f a, v2f b, v8f c) {
  // V_WMMA_F32_16X16X4_F32: exact fp32 matrix op, D = A(16x4) x B(4x16) + C
  return __builtin_amdgcn_wmma_f32_16x16x4_f32(
      /*neg_a=*/false, a, /*neg_b=*/false, b,
      /*c_mod=*/(short)0, c, /*reuse_a=*/false, /*reuse_b=*/false);
}

__device__ __forceinline__ float gelu_tanh(float x) {
  // jax.nn.gelu default (approximate=True)
  const float k0 = 0.7978845608028654f;
  const float k1 = 0.044715f;
  float x3 = x * x * x;
  return 0.5f * x * (1.0f + tanhf(k0 * (x + k1 * x3)));
}

// ---------------------------------------------------------------------------
// Kernel A: per-node first-layer Khatri-Rao einsums.
// grid = 625 tiles * 20 jobs, block = 1 wave (32 lanes).
// jobs 0-3: nf_s col-tiles (K=1024), 4-7: self_s,
// jobs 8-13: nf_v (comp,tile) (K=512), 14-19: self_v.
// ---------------------------------------------------------------------------
__global__ __launch_bounds__(32) void node_pre_kernel(
    const float* __restrict__ node_input, const float* __restrict__ node_attr,
    const float* __restrict__ W1a, const float* __restrict__ W1b,
    const float* __restrict__ W1c, const float* __restrict__ W1d,
    float* __restrict__ nf_s, float* __restrict__ self_s,
    float* __restrict__ nf_v, float* __restrict__ self_v) {
  __shared__ float act[16 * 97];  // s path uses stride 65, v path stride 97
  __shared__ float at[16 * 17];
  int lane = threadIdx.x;
  int job = blockIdx.x % 20;
  int tile = blockIdx.x / 20;
  int base = tile * 16;

  for (int idx = lane; idx < 256; idx += 32) {
    int r = idx >> 4, c = idx & 15;
    at[r * 17 + c] = node_attr[(base + r) * 16 + c];
  }
  bool is_s = (job < 8);
  if (is_s) {
    for (int idx = lane; idx < 16 * 64; idx += 32) {
      int r = idx >> 6, c = idx & 63;
      act[r * 65 + c] = node_input[(base + r) * 160 + c];
    }
  } else {
    for (int idx = lane; idx < 16 * 96; idx += 32) {
      int r = idx / 96, c = idx % 96;
      act[r * 97 + c] = node_input[(base + r) * 160 + 64 + c];
    }
  }
  __syncthreads();

  int m = lane & 15, half = lane >> 4, n = lane & 15;
  v8f acc = {};
  if (is_s) {
    const float* W = (job < 4) ? W1a : W1b;  // (64,16,64) -> flat (1024 x 64)
    int wt = job & 3;
    const float* Bp = W + wt * 16 + n;
    for (int kk = 0; kk < 256; ++kk) {
      int K0 = kk * 4 + half * 2;
      int K1 = K0 + 1;
      v2f a, b;
      a[0] = act[m * 65 + (K0 >> 4)] * at[m * 17 + (K0 & 15)];
      a[1] = act[m * 65 + (K1 >> 4)] * at[m * 17 + (K1 & 15)];
      b[0] = Bp[K0 * 64];
      b[1] = Bp[K1 * 64];
      acc = wmma4(a, b, acc);
    }
    const float scale = 0.03125f;  // 1/sqrt(64*16)
    float* out = (job < 4) ? nf_s : self_s;
#pragma unroll
    for (int r = 0; r < 8; ++r) {
      int row = r + 8 * half;
      out[(base + row) * 64 + wt * 16 + n] = acc[r] * scale;
    }
  } else {
    int j = job - 8;
    bool isnf = (j < 6);
    int jj = isnf ? j : j - 6;
    int comp = jj >> 1, wt = jj & 1;
    const float* W = isnf ? W1c : W1d;  // (32,16,32) -> flat (512 x 32)
    const float* Bp = W + wt * 16 + n;
    for (int kk = 0; kk < 128; ++kk) {
      int K0 = kk * 4 + half * 2;
      int K1 = K0 + 1;
      v2f a, b;
      a[0] = act[m * 97 + (K0 >> 4) * 3 + comp] * at[m * 17 + (K0 & 15)];
      a[1] = act[m * 97 + (K1 >> 4) * 3 + comp] * at[m * 17 + (K1 & 15)];
      b[0] = Bp[K0 * 32];
      b[1] = Bp[K1 * 32];
      acc = wmma4(a, b, acc);
    }
    const float scale = 0.044194173824159216f;  // 1/sqrt(32*16)
    float* out = isnf ? nf_v : self_v;
#pragma unroll
    for (int r = 0; r < 8; ++r) {
      int row = r + 8 * half;
      out[(base + row) * 96 + (wt * 16 + n) * 3 + comp] = acc[r] * scale;
    }
  }
}

// ---------------------------------------------------------------------------
// Kernel B: fused edge pipeline. 1 wave handles 16 edges:
// MLP (WMMA) -> w0..w3 (WMMA) -> gathers -> tensor products -> atomic scatter.
// Nothing edge-sized ever touches HBM except the L2-resident accumulator.
// ---------------------------------------------------------------------------
__global__ __launch_bounds__(32) void edge_kernel(
    const float* __restrict__ nf_s, const float* __restrict__ nf_v,
    const int* __restrict__ edge_src, const int* __restrict__ edge_dst,
    const float* __restrict__ edge_attr,
    const float* __restrict__ edge_scalar_attr,
    const float* __restrict__ M1, const float* __restrict__ M2,
    const float* __restrict__ P0, const float* __restrict__ P1,
    const float* __restrict__ P2, const float* __restrict__ P3,
    float* __restrict__ agg) {
  __shared__ float esa[16 * 9];
  __shared__ float hbuf[16 * 65];
  __shared__ float h2buf[16 * 65];
  __shared__ float wbuf[16 * 193];  // cols: [0,64)=w0 [64,128)=w1 [128,160)=w2 [160,192)=w3
  __shared__ float sebuf[16 * 65];
  __shared__ float vebuf[16 * 97];
  __shared__ float e0buf[16];
  __shared__ float e1buf[16 * 3];
  __shared__ int dstbuf[16];
  __shared__ int srcbuf[16];

  int lane = threadIdx.x;
  int ebase = blockIdx.x * 16;
  if (lane < 16) {
    srcbuf[lane] = edge_src[ebase + lane];
    dstbuf[lane] = edge_dst[ebase + lane];
    e0buf[lane] = edge_attr[(ebase + lane) * 4 + 0];
  }
  for (int idx = lane; idx < 16 * 3; idx += 32) {
    int r = idx / 3, c = idx - r * 3;
    e1buf[idx] = edge_attr[(ebase + r) * 4 + 1 + c];
  }
  for (int idx = lane; idx < 16 * 8; idx += 32) {
    int r = idx >> 3, c = idx & 7;
    esa[r * 9 + c] = edge_scalar_attr[(ebase + r) * 8 + c];
  }
  __syncthreads();
  // gathers of node features along edge_src
  for (int idx = lane; idx < 16 * 64; idx += 32) {
    int r = idx >> 6, c = idx & 63;
    sebuf[r * 65 + c] = nf_s[srcbuf[r] * 64 + c];
  }
  for (int idx = lane; idx < 16 * 96; idx += 32) {
    int r = idx / 96, c = idx % 96;
    vebuf[r * 97 + c] = nf_v[srcbuf[r] * 96 + c];
  }

  int m = lane & 15, half = lane >> 4, n = lane & 15;
  // h = gelu(esa @ M1 / sqrt(8))
  const float rs8 = 0.35355339059327373f;
  for (int t = 0; t < 4; ++t) {
    v8f acc = {};
    for (int kk = 0; kk < 2; ++kk) {
      int K0 = kk * 4 + half * 2, K1 = K0 + 1;
      v2f a, b;
      a[0] = esa[m * 9 + K0];
      a[1] = esa[m * 9 + K1];
      b[0] = M1[K0 * 64 + t * 16 + n];
      b[1] = M1[K1 * 64 + t * 16 + n];
      acc = wmma4(a, b, acc);
    }
#pragma unroll
    for (int r = 0; r < 8; ++r) {
      int row = r + 8 * half;
      hbuf[row * 65 + t * 16 + n] = gelu_tanh(acc[r] * rs8);
    }
  }
  __syncthreads();
  // h2 = gelu(h @ M2 / 8)
  for (int t = 0; t < 4; ++t) {
    v8f acc = {};
    for (int kk = 0; kk < 16; ++kk) {
      int K0 = kk * 4 + half * 2, K1 = K0 + 1;
      v2f a, b;
      a[0] = hbuf[m * 65 + K0];
      a[1] = hbuf[m * 65 + K1];
      b[0] = M2[K0 * 64 + t * 16 + n];
      b[1] = M2[K1 * 64 + t * 16 + n];
      acc = wmma4(a, b, acc);
    }
#pragma unroll
    for (int r = 0; r < 8; ++r) {
      int row = r + 8 * half;
      h2buf[row * 65 + t * 16 + n] = gelu_tanh(acc[r] * 0.125f);
    }
  }
  __syncthreads();
  // w = h2 @ [P0 | P1 | P2 | P3]  (64 x 192)
  for (int t = 0; t < 12; ++t) {
    const float* P;
    int col, ldb;
    if (t < 4)       { P = P0; col = t * 16;        ldb = 64; }
    else if (t < 8)  { P = P1; col = (t - 4) * 16;  ldb = 64; }
    else if (t < 10) { P = P2; col = (t - 8) * 16;  ldb = 32; }
    else             { P = P3; col = (t - 10) * 16; ldb = 32; }
    v8f acc = {};
    for (int kk = 0; kk < 16; ++kk) {
      int K0 = kk * 4 + half * 2, K1 = K0 + 1;
      v2f a, b;
      a[0] = h2buf[m * 65 + K0];
      a[1] = h2buf[m * 65 + K1];
      b[0] = P[K0 * ldb + col + n];
      b[1] = P[K1 * ldb + col + n];
      acc = wmma4(a, b, acc);
    }
#pragma unroll
    for (int r = 0; r < 8; ++r) {
      int row = r + 8 * half;
      wbuf[row * 193 + t * 16 + n] = acc[r];
    }
  }
  __syncthreads();
  // tensor products + scatter-add (L2-resident accumulator)
  const float is3 = 0.5773502691896258f;
  for (int idx = lane; idx < 16 * 384; idx += 32) {
    int e = idx / 384, f = idx - e * 384;
    const float* w = &wbuf[e * 193];
    float val;
    if (f < 64) {                       // out0a = w0*se*e0
      val = w[f] * sebuf[e * 65 + f] * e0buf[e];
    } else if (f < 96) {                // out0b = w3*(ve.e1)/sqrt3
      int u = f - 64;
      float dot = vebuf[e * 97 + u * 3 + 0] * e1buf[e * 3 + 0] +
                  vebuf[e * 97 + u * 3 + 1] * e1buf[e * 3 + 1] +
                  vebuf[e * 97 + u * 3 + 2] * e1buf[e * 3 + 2];
      val = w[160 + u] * dot * is3;
    } else if (f < 288) {               // out1a = (w1*se) x e1
      int g = f - 96;
      int u = g / 3, i = g - u * 3;
      val = w[64 + u] * sebuf[e * 65 + u] * e1buf[e * 3 + i];
    } else {                            // out1b = w2*ve*e0
      int g = f - 288;
      int u = g / 3, i = g - u * 3;
      val = w[128 + u] * vebuf[e * 97 + u * 3 + i] * e0buf[e];
    }
    atomicAdd(&agg[dstbuf[e] * 384 + f], val);
  }
}

// ---------------------------------------------------------------------------
// Kernel C: second-layer einsums (K=1536) + mix + output concat.
// grid = 625 tiles * 10 jobs. jobs 0-3: conv_s tiles, 4-9: conv_v (comp,tile).
// ---------------------------------------------------------------------------
__global__ __launch_bounds__(32) void node_post_kernel(
    const float* __restrict__ agg, const float* __restrict__ node_attr,
    const float* __restrict__ W2s, const float* __restrict__ W2v,
    const float* __restrict__ self_s, const float* __restrict__ self_v,
    float* __restrict__ out) {
  __shared__ float mid[16 * 289];  // s path stride 97, v path stride 289
  __shared__ float at[16 * 17];
  int lane = threadIdx.x;
  int job = blockIdx.x % 10;
  int tile = blockIdx.x / 10;
  int base = tile * 16;
  const float inv_sqrt_nn = 0.1767766952966369f;  // 1/sqrt(32)

  for (int idx = lane; idx < 256; idx += 32) {
    int r = idx >> 4, c = idx & 15;
    at[r * 17 + c] = node_attr[(base + r) * 16 + c];
  }
  bool is_s = (job < 4);
  if (is_s) {
    for (int idx = lane; idx < 16 * 96; idx += 32) {
      int r = idx / 96, c = idx % 96;
      mid[r * 97 + c] = agg[(base + r) * 384 + c] * inv_sqrt_nn;
    }
  } else {
    for (int idx = lane; idx < 16 * 288; idx += 32) {
      int r = idx / 288, c = idx % 288;
      mid[r * 289 + c] = agg[(base + r) * 384 + 96 + c] * inv_sqrt_nn;
    }
  }
  __syncthreads();

  int m = lane & 15, half = lane >> 4, n = lane & 15;
  const float n2 = 0.02551551815399144f;  // 1/sqrt(96*16)
  const float cmix = 0.9800665778412416f; // cos(0.2)
  const float smix = 0.19866933079506122f;// sin(0.2)
  v8f acc = {};
  if (is_s) {
    int wt = job;
    const float* Bp = W2s + wt * 16 + n;  // (96,16,64) -> flat (1536 x 64)
    for (int kk = 0; kk < 384; ++kk) {
      int K0 = kk * 4 + half * 2, K1 = K0 + 1;
      v2f a, b;
      a[0] = mid[m * 97 + (K0 >> 4)] * at[m * 17 + (K0 & 15)];
      a[1] = mid[m * 97 + (K1 >> 4)] * at[m * 17 + (K1 & 15)];
      b[0] = Bp[K0 * 64];
      b[1] = Bp[K1 * 64];
      acc = wmma4(a, b, acc);
    }
#pragma unroll
    for (int r = 0; r < 8; ++r) {
      int node = base + r + 8 * half;
      int col = wt * 16 + n;
      out[node * 160 + col] =
          cmix * self_s[node * 64 + col] + smix * (acc[r] * n2);
    }
  } else {
    int j = job - 4;
    int comp = j >> 1, wt = j & 1;
    const float* Bp = W2v + wt * 16 + n;  // (96,16,32) -> flat (1536 x 32)
    for (int kk = 0; kk < 384; ++kk) {
      int K0 = kk * 4 + half * 2, K1 = K0 + 1;
      v2f a, b;
      a[0] = mid[m * 289 + (K0 >> 4) * 3 + comp] * at[m * 17 + (K0 & 15)];
      a[1] = mid[m * 289 + (K1 >> 4) * 3 + comp] * at[m * 17 + (K1 & 15)];
      b[0] = Bp[K0 * 32];
      b[1] = Bp[K1 * 32];
      acc = wmma4(a, b, acc);
    }
#pragma unroll
    for (int r = 0; r < 8; ++r) {
      int node = base + r + 8 * half;
      int w = wt * 16 + n;
      out[node * 160 + 64 + w * 3 + comp] =
          cmix * self_v[node * 96 + w * 3 + comp] + smix * (acc[r] * n2);
    }
  }
}

extern "C" void kernel_launch(void* const* d_in, const int* in_sizes, int n_in,
                              void* d_out, int out_size, void* d_ws,
                              size_t ws_size, hipStream_t stream) {
  const float* node_input  = (const float*)d_in[0];
  const float* node_attr   = (const float*)d_in[1];
  const int*   edge_src    = (const int*)d_in[2];
  const int*   edge_dst    = (const int*)d_in[3];
  const float* edge_attr   = (const float*)d_in[4];
  const float* edge_scalar = (const float*)d_in[5];
  const float* W1a = (const float*)d_in[6];
  const float* W1b = (const float*)d_in[7];
  const float* W1c = (const float*)d_in[8];
  const float* W1d = (const float*)d_in[9];
  const float* M1  = (const float*)d_in[10];
  const float* M2  = (const float*)d_in[11];
  const float* P0  = (const float*)d_in[12];
  const float* P1  = (const float*)d_in[13];
  const float* P2  = (const float*)d_in[14];
  const float* P3  = (const float*)d_in[15];
  const float* W2s = (const float*)d_in[16];
  const float* W2v = (const float*)d_in[17];
  float* out = (float*)d_out;
  float* ws = (float*)d_ws;

  float* nf_s   = ws;                   // N*64
  float* self_s = ws + N_NODES * 64;    // N*64
  float* nf_v   = ws + N_NODES * 128;   // N*96
  float* self_v = ws + N_NODES * 224;   // N*96
  float* agg    = ws + N_NODES * 320;   // N*384

  hipMemsetAsync(agg, 0, (size_t)N_NODES * 384 * sizeof(float), stream);

  dim3 blk(32);
  node_pre_kernel<<<dim3((N_NODES / 16) * 20), blk, 0, stream>>>(
      node_input, node_attr, W1a, W1b, W1c, W1d, nf_s, self_s, nf_v, self_v);
  edge_kernel<<<dim3(N_EDGES / 16), blk, 0, stream>>>(
      nf_s, nf_v, edge_src, edge_dst, edge_attr, edge_scalar, M1, M2, P0, P1,
      P2, P3, agg);
  node_post_kernel<<<dim3((N_NODES / 16) * 10), blk, 0, stream>>>(
      agg, node_attr, W2s, W2v, self_s, self_v, out);
}